// TimeKernelAttentionA_61125974557079
// MI455X (gfx1250) — compile-verified
//
#include <hip/hip_runtime.h>

// ---------------------------------------------------------------------------
// Problem constants (match reference): B=2, L=256, DM=256, H=8, NK=4, T=4
// ---------------------------------------------------------------------------
constexpr int Bc  = 2;
constexpr int Lc  = 256;
constexpr int DMc = 256;
constexpr int Hc  = 8;
constexpr int NKc = 4;
constexpr int Tc  = 4;
constexpr int DHc = 32;              // DM / H
constexpr int PAR = DHc * Tc * NKc;  // 512 gaussian params per head
constexpr int Mc  = Bc * Lc;         // 512 rows in the projection GEMMs

#define LOG2E 1.44269504088896340736f

typedef __attribute__((ext_vector_type(2))) float v2f;
typedef __attribute__((ext_vector_type(8))) float v8f;

// CDNA5 full-precision WMMA: D(16x16,f32) = A(16x4,f32) * B(4x16,f32) + C
__device__ __forceinline__ v8f wmma_f32_k4(v2f a, v2f b, v8f c) {
  return __builtin_amdgcn_wmma_f32_16x16x4_f32(
      /*neg_a=*/false, a, /*neg_b=*/false, b,
      /*c_mod=*/(short)0, c, /*reuse_a=*/false, /*reuse_b=*/false);
}

// ---------------------------------------------------------------------------
// C[512,256] = A[512,256] * B[256,256]^T (+ bias[256]).  All dims constexpr so
// every load is an immediate-offset access off two base registers -> the
// clause-former batches loads and emits graduated s_wait_loadcnt instead of
// full drains before each WMMA.
// One wave per 16(M) x 64(N) strip: A-frag loaded once, reused for 4 WMMAs.
// Frag layouts per ISA 7.12.2:
//   A: a.x = A[m0+lo][kk+2*hi], a.y = +1
//   B: b.x = B[n0+lo][kk+2*hi], b.y = +1
//   D: VGPR r -> element (m0 + r + 8*hi, n0 + lo)
// ---------------------------------------------------------------------------
__global__ void gemm_abt(const float* __restrict__ A, const float* __restrict__ B,
                         const float* __restrict__ bias, float* __restrict__ C) {
  const int lane = threadIdx.x & 31;
  const int wave = threadIdx.x >> 5;
  constexpr int NTG = DMc / 64;                    // 4 groups of 4 N-tiles
  const int unit = blockIdx.x * (blockDim.x >> 5) + wave;
  if (unit >= (Mc / 16) * NTG) return;             // wave-uniform guard
  const int mt = unit / NTG, ng = unit % NTG;
  const int lo = lane & 15, hi = lane >> 4;

  const float* arow = A + (size_t)(mt * 16 + lo) * DMc + 2 * hi;
  const float* brow = B + (size_t)(ng * 64 + lo) * DMc + 2 * hi;

  v8f acc[4] = {};
#pragma unroll 8
  for (int kk = 0; kk < DMc; kk += 4) {
    const v2f a  = *(const v2f*)(arow + kk);
    const v2f b0 = *(const v2f*)(brow + kk);
    const v2f b1 = *(const v2f*)(brow + 16 * DMc + kk);
    const v2f b2 = *(const v2f*)(brow + 32 * DMc + kk);
    const v2f b3 = *(const v2f*)(brow + 48 * DMc + kk);
    acc[0] = wmma_f32_k4(a, b0, acc[0]);
    acc[1] = wmma_f32_k4(a, b1, acc[1]);
    acc[2] = wmma_f32_k4(a, b2, acc[2]);
    acc[3] = wmma_f32_k4(a, b3, acc[3]);
  }
#pragma unroll
  for (int s = 0; s < 4; ++s) {
    const int col = ng * 64 + s * 16 + lo;
    const float bv = bias ? bias[col] : 0.0f;
#pragma unroll
    for (int r = 0; r < 8; ++r) {
      const int row = mt * 16 + r + 8 * hi;
      C[(size_t)row * DMc + col] = acc[s][r] + bv;
    }
  }
}

// ---------------------------------------------------------------------------
// g_abs[b,h,i] = (1/DH) * sum_{d,t,k} w*exp(-(t[b,i,t]-mu)^2/(2 sigma^2))
// One block per (b,h); thread == i.  Params staged in LDS (coef premul log2e).
// ---------------------------------------------------------------------------
__global__ void gauss_abs_kernel(const float* __restrict__ t,
                                 const float* __restrict__ mu,
                                 const float* __restrict__ sigma,
                                 const float* __restrict__ w,
                                 float* __restrict__ g) {
  __shared__ float smu[PAR], scf[PAR], swt[PAR];
  const int bh = blockIdx.x;          // b*H + h
  const int b = bh / Hc, h = bh % Hc;
  const int tid = threadIdx.x;
  for (int s = tid; s < PAR; s += blockDim.x) {
    smu[s] = mu[h * PAR + s];
    const float sg = sigma[h * PAR + s];
    scf[s] = LOG2E / (2.0f * sg * sg);
    swt[s] = w[h * PAR + s];
  }
  __syncthreads();

  const int i = tid;                  // L == blockDim.x == 256
  float ti[Tc];
#pragma unroll
  for (int tt = 0; tt < Tc; ++tt) ti[tt] = t[(b * Lc + i) * Tc + tt];

  float acc = 0.0f;
  for (int d = 0; d < DHc; ++d) {
#pragma unroll
    for (int tt = 0; tt < Tc; ++tt) {
#pragma unroll
      for (int k = 0; k < NKc; ++k) {
        const int idx = (d * Tc + tt) * NKc + k;
        const float df = ti[tt] - smu[idx];
        acc += swt[idx] * __builtin_amdgcn_exp2f(-df * df * scf[idx]);
      }
    }
  }
  g[bh * Lc + i] = acc * (1.0f / DHc);
}

// ---------------------------------------------------------------------------
// g_rel[b,h,i,j] over u = |t_i - t_j|.  SYMMETRIC in (i,j): compute the upper
// triangle only and mirror -> halves the 537M-exp hotspot.  One block per
// (b,h,i); thread == j; lanes with j < i retire before the loop, so waves
// entirely below the diagonal do no transcendental work.
// ---------------------------------------------------------------------------
__global__ void gauss_rel_kernel(const float* __restrict__ t,
                                 const float* __restrict__ mu,
                                 const float* __restrict__ sigma,
                                 const float* __restrict__ w,
                                 float* __restrict__ g) {
  __shared__ float smu[PAR], scf[PAR], swt[PAR];
  const int blk = blockIdx.x;         // bh*L + i
  const int bh = blk >> 8;
  const int i  = blk & (Lc - 1);
  const int b = bh / Hc, h = bh % Hc;
  const int tid = threadIdx.x;
  for (int s = tid; s < PAR; s += blockDim.x) {
    smu[s] = mu[h * PAR + s];
    const float sg = sigma[h * PAR + s];
    scf[s] = LOG2E / (2.0f * sg * sg);
    swt[s] = w[h * PAR + s];
  }
  __syncthreads();                    // no barriers after this point

  const int j = tid;
  if (j < i) return;                  // upper triangle only

  float u[Tc];
#pragma unroll
  for (int tt = 0; tt < Tc; ++tt)
    u[tt] = fabsf(t[(b * Lc + i) * Tc + tt] - t[(b * Lc + j) * Tc + tt]);

  float acc = 0.0f;
  for (int d = 0; d < DHc; ++d) {
#pragma unroll
    for (int tt = 0; tt < Tc; ++tt) {
#pragma unroll
      for (int k = 0; k < NKc; ++k) {
        const int idx = (d * Tc + tt) * NKc + k;
        const float df = u[tt] - smu[idx];
        acc += swt[idx] * __builtin_amdgcn_exp2f(-df * df * scf[idx]);
      }
    }
  }
  const float r = acc * (1.0f / DHc);
  g[((size_t)bh * Lc + i) * Lc + j] = r;
  g[((size_t)bh * Lc + j) * Lc + i] = r;   // mirror (diagonal rewrites same value)
}

// ---------------------------------------------------------------------------
// logits[b,h,i,j] = (Q.K^T/sqrt(DH)) * (alpha*P*2*g_abs_i + beta*P*g_rel + gamma)
// Fuses the scores GEMM and the Pij = Xh.Xh^T GEMM (both WMMA over K=32).
// One wave per 16x16 (i,j) tile; tiles = B*H * 16 * 16 = 4096.
// ---------------------------------------------------------------------------
__global__ void logits_kernel(const float* __restrict__ Q, const float* __restrict__ Km,
                              const float* __restrict__ x,
                              const float* __restrict__ g_abs, const float* __restrict__ g_rel,
                              const float* __restrict__ alpha, const float* __restrict__ beta,
                              const float* __restrict__ gamma,
                              float* __restrict__ logits) {
  const int lane = threadIdx.x & 31;
  const int wave = threadIdx.x >> 5;
  const int tile = blockIdx.x * (blockDim.x >> 5) + wave;
  const int nt = tile & 15;
  const int mt = (tile >> 4) & 15;
  const int bh = tile >> 8;
  const int b = bh >> 3, h = bh & 7;
  const int lo = lane & 15, hi = lane >> 4;

  const float* qrow = Q  + (size_t)(b * Lc + mt * 16 + lo) * DMc + h * DHc + 2 * hi;
  const float* krow = Km + (size_t)(b * Lc + nt * 16 + lo) * DMc + h * DHc + 2 * hi;
  const float* xi   = x  + (size_t)(b * Lc + mt * 16 + lo) * DMc + h * DHc + 2 * hi;
  const float* xj   = x  + (size_t)(b * Lc + nt * 16 + lo) * DMc + h * DHc + 2 * hi;

  v8f cS = {}, cP = {};
#pragma unroll
  for (int kk = 0; kk < DHc; kk += 4) {
    cS = wmma_f32_k4(*(const v2f*)(qrow + kk), *(const v2f*)(krow + kk), cS);
    cP = wmma_f32_k4(*(const v2f*)(xi + kk), *(const v2f*)(xj + kk), cP);
  }

  const float al = alpha[h], be = beta[h], ga = gamma[h];
  const float invs = 0.17677669529663688f;   // 1/sqrt(32)
#pragma unroll
  for (int r = 0; r < 8; ++r) {
    const int i = mt * 16 + r + 8 * hi;
    const int j = nt * 16 + lo;
    const float gi = g_abs[bh * Lc + i];
    const float gr = g_rel[((size_t)bh * Lc + i) * Lc + j];
    const float Av = al * cP[r] * (2.0f * gi) + be * cP[r] * gr + ga;
    logits[((size_t)bh * Lc + i) * Lc + j] = cS[r] * invs * Av;
  }
}

// ---------------------------------------------------------------------------
// In-place row softmax over 256 columns. One 256-thread block per row.
// ---------------------------------------------------------------------------
__global__ void softmax_rows(float* __restrict__ buf) {
  __shared__ float red[256];
  const int row = blockIdx.x;
  const int tid = threadIdx.x;
  const float v = buf[(size_t)row * Lc + tid];
  red[tid] = v;
  __syncthreads();
  for (int s = 128; s > 0; s >>= 1) {
    if (tid < s) red[tid] = fmaxf(red[tid], red[tid + s]);
    __syncthreads();
  }
  const float m = red[0];
  __syncthreads();
  const float e = __builtin_amdgcn_exp2f((v - m) * LOG2E);
  red[tid] = e;
  __syncthreads();
  for (int s = 128; s > 0; s >>= 1) {
    if (tid < s) red[tid] += red[tid + s];
    __syncthreads();
  }
  buf[(size_t)row * Lc + tid] = e / red[0];
}

// ---------------------------------------------------------------------------
// ctx[b, i, h*DH + d] = sum_j attn[b,h,i,j] * V[b, j, h*DH + d]
// One wave per 16(M) x 32(DH) strip: attn fragment reused across both N-tiles.
// B is dense row-major [K=256, N=32]: b.x = V[kk+2*hi][n], b.y = next K row.
// All strides constexpr -> immediate-offset loads off two base registers.
// Wave-units = B*H * (L/16) = 256 -> 32 blocks of 8 waves.
// ---------------------------------------------------------------------------
__global__ void attnv_kernel(const float* __restrict__ attn, const float* __restrict__ V,
                             float* __restrict__ ctx) {
  const int lane = threadIdx.x & 31;
  const int wave = threadIdx.x >> 5;
  const int unit = blockIdx.x * (blockDim.x >> 5) + wave;
  const int mt = unit & 15;
  const int bh = unit >> 4;
  const int b = bh >> 3, h = bh & 7;
  const int lo = lane & 15, hi = lane >> 4;

  const float* arow = attn + ((size_t)bh * Lc + mt * 16 + lo) * Lc + 2 * hi;
  const float* bcol = V + (size_t)(b * Lc + 2 * hi) * DMc + h * DHc + lo;

  v8f c0 = {}, c1 = {};
#pragma unroll 8
  for (int kk = 0; kk < Lc; kk += 4) {
    const v2f a = *(const v2f*)(arow + kk);
    v2f b0, b1;
    b0.x = bcol[kk * DMc];
    b0.y = bcol[(kk + 1) * DMc];
    b1.x = bcol[kk * DMc + 16];
    b1.y = bcol[(kk + 1) * DMc + 16];
    c0 = wmma_f32_k4(a, b0, c0);
    c1 = wmma_f32_k4(a, b1, c1);
  }
#pragma unroll
  for (int r = 0; r < 8; ++r) {
    const int i = mt * 16 + r + 8 * hi;
    float* dst = ctx + (size_t)(b * Lc + i) * DMc + h * DHc + lo;
    dst[0]  = c0[r];
    dst[16] = c1[r];
  }
}

// ---------------------------------------------------------------------------
// Host-side orchestration
// ---------------------------------------------------------------------------
extern "C" void kernel_launch(void* const* d_in, const int* in_sizes, int n_in,
                              void* d_out, int out_size, void* d_ws, size_t ws_size,
                              hipStream_t stream) {
  (void)in_sizes; (void)n_in; (void)out_size; (void)ws_size;
  const float* x        = (const float*)d_in[0];
  const float* t        = (const float*)d_in[1];
  const float* WQ       = (const float*)d_in[2];
  const float* WK       = (const float*)d_in[3];
  const float* WV       = (const float*)d_in[4];
  const float* WO       = (const float*)d_in[5];
  const float* bO       = (const float*)d_in[6];
  const float* mu_abs   = (const float*)d_in[7];
  const float* sig_abs  = (const float*)d_in[8];
  const float* w_abs    = (const float*)d_in[9];
  const float* mu_rel   = (const float*)d_in[10];
  const float* sig_rel  = (const float*)d_in[11];
  const float* w_rel    = (const float*)d_in[12];
  const float* alpha    = (const float*)d_in[13];
  const float* beta     = (const float*)d_in[14];
  const float* gamma    = (const float*)d_in[15];

  float* ws = (float*)d_ws;
  float* Q      = ws;                               // 131072
  float* Kb     = ws + 131072;                      // 131072
  float* Vb     = ws + 262144;                      // 131072
  float* ctx    = ws + 393216;                      // 131072
  float* g_abs  = ws + 524288;                      // 4096
  float* g_rel  = ws + 528384;                      // 1048576
  float* logits = ws + 1576960;                     // 1048576  (-> attn in place)

  // Q/K/V projections: 128 wave-units each (16x64 strips) -> 16 blocks
  gemm_abt<<<16, 256, 0, stream>>>(x, WQ, nullptr, Q);
  gemm_abt<<<16, 256, 0, stream>>>(x, WK, nullptr, Kb);
  gemm_abt<<<16, 256, 0, stream>>>(x, WV, nullptr, Vb);

  // Gaussian biases (g_rel: symmetric upper-triangle + mirror)
  gauss_abs_kernel<<<Bc * Hc, 256, 0, stream>>>(t, mu_abs, sig_abs, w_abs, g_abs);
  gauss_rel_kernel<<<Bc * Hc * Lc, 256, 0, stream>>>(t, mu_rel, sig_rel, w_rel, g_rel);

  // Fused scores + Pij + combine: 4096 tiles / 8 waves -> 512 blocks
  logits_kernel<<<512, 256, 0, stream>>>(Q, Kb, x, g_abs, g_rel,
                                         alpha, beta, gamma, logits);

  // Row softmax: one block per (b,h,i) row
  softmax_rows<<<Bc * Hc * Lc, 256, 0, stream>>>(logits);

  // attn @ V: 256 wave-units -> 32 blocks; ctx written in [B, L, H*DH] layout
  attnv_kernel<<<32, 256, 0, stream>>>(logits, Vb, ctx);

  // Final projection + bias
  gemm_abt<<<16, 256, 0, stream>>>(ctx, WO, bO, (float*)d_out);
}